// WaveRNN_56272661512470
// MI455X (gfx1250) — compile-verified
//
#include <hip/hip_runtime.h>
#include <stdint.h>

// Problem constants (match reference)
#define B_     4
#define NT_    256
#define NZ_    384
#define NX_    384
#define NREC_  128
#define NZP_   386                      // +2 guard rows (top/bottom), stay zero
#define DT_    0.001f

#define ROWS_PER_WG   16
#define WG_PER_B      24                // 24*16 = 384 rows
#define NWG           (B_ * WG_PER_B)   // 96 workgroups (persistent, co-resident)
#define NTHREADS      256
#define TILE_ROWS     (ROWS_PER_WG + 2) // 18 rows incl. halo
#define PTS_PER_WG    (ROWS_PER_WG * NX_)      // 6144
#define PTS_PER_THREAD (PTS_PER_WG / NTHREADS) // 24

typedef uint32_t u32x4 __attribute__((ext_vector_type(4)));
typedef int32_t  i32x4 __attribute__((ext_vector_type(4)));
typedef int32_t  i32x8 __attribute__((ext_vector_type(8)));

#if defined(__HIP_DEVICE_COMPILE__) && defined(__has_builtin)
#if __has_builtin(__builtin_amdgcn_tensor_load_to_lds)
#define HAVE_TDM 1
#endif
#endif

#ifdef HAVE_TDM
// Issue one TDM DMA: 18x384 f32 tile, global (L2) -> LDS. Caller gates to one wave.
__device__ __forceinline__ void tdm_issue_tile(unsigned lds_off, uint64_t ga) {
    u32x4 g0;
    g0[0] = 1u;                                    // count=1, user descriptor
    g0[1] = lds_off;                               // lds_addr (bytes)
    g0[2] = (uint32_t)ga;                          // global_addr lo32
    g0[3] = (uint32_t)((ga >> 32) & 0x1FFFFFFu) | (2u << 30); // addr hi | type=2
    i32x8 g1;
    g1[0] = (int)(2u << 16);                       // data_size = 4 bytes
    g1[1] = (int)((unsigned)NX_ << 16);            // tensor_dim0[15:0] = 384
    g1[2] = (int)((unsigned)TILE_ROWS << 16);      // tensor_dim1[15:0] = 18
    g1[3] = (int)((unsigned)NX_ << 16);            // tile_dim0 = 384
    g1[4] = TILE_ROWS;                             // tile_dim1 = 18
    g1[5] = NX_;                                   // tensor_dim0_stride = 384
    g1[6] = 0; g1[7] = 0;
    i32x4 gz = {0, 0, 0, 0};
#if __clang_major__ >= 23
    i32x8 gz8 = {0, 0, 0, 0, 0, 0, 0, 0};
    __builtin_amdgcn_tensor_load_to_lds(g0, g1, gz, gz, gz8, 16 /*scope=DEV*/);
#else
    __builtin_amdgcn_tensor_load_to_lds(g0, g1, gz, gz, 16 /*scope=DEV*/);
#endif
}
#endif

__global__ __launch_bounds__(NTHREADS)
void wave_persist_kernel(const float* __restrict__ xsrc,     // (B, NT)
                         const float* __restrict__ vp,       // (NZ, NX)
                         const int*   __restrict__ src_y,
                         const int*   __restrict__ src_x,
                         const int*   __restrict__ rec_y,
                         const int*   __restrict__ rec_x,
                         float*       __restrict__ out,      // (NT, B, NREC)
                         float* bufA, float* bufB, unsigned* barrier_cnt)
{
    __shared__ float lds_h1[TILE_ROWS * NX_];       // 27648 B: h1 tile incl. halo
    __shared__ float lds_c2[ROWS_PER_WG * NX_];     // 24576 B: (vp*DT)^2, time-invariant

    const int wg  = blockIdx.x;
    const int tid = threadIdx.x;
    const int b   = wg / WG_PER_B;
    const int z0  = (wg % WG_PER_B) * ROWS_PER_WG;  // first real row owned by this WG
    const size_t batch_base = (size_t)b * NZP_ * NX_;
    const size_t tile_off   = batch_base + (size_t)z0 * NX_;  // padded row z0 = real z0-1
    const unsigned lds_off  = (unsigned)(uintptr_t)(void*)&lds_h1[0];

#ifdef HAVE_TDM
    // Prologue: kick off the t=0 tile DMA (bufA, all zeros) so it overlaps with
    // the c2dt2 staging below. TDM ignores EXEC; gate to wave 0 only.
    if (tid < 32)
        tdm_issue_tile(lds_off, (uint64_t)(uintptr_t)(bufA + tile_off));
#endif

    // Stage c2dt2 = (vp*DT)^2 for our rows once (matches reference fp order).
    for (int j = tid; j < PTS_PER_WG; j += NTHREADS) {
        int r = j / NX_, c = j - r * NX_;
        float a = vp[(size_t)(z0 + r) * NX_ + c] * DT_;
        lds_c2[j] = a * a;
    }

    const int sy = src_y[b];
    const int sx = src_x[b];
    const float inv_dh2 = 0.01f;                    // 1/(DH*DH)

    // Receiver gather assignment: first B*NREC (=512) global threads.
    const int gtid = wg * NTHREADS + tid;
    size_t goff = 0;
    if (gtid < B_ * NREC_) {
        int gb = gtid >> 7, gr = gtid & 127;
        goff = (size_t)gb * NZP_ * NX_ + (size_t)(rec_y[gr] + 1) * NX_ + rec_x[gr];
    }
    __syncthreads();   // lds_c2 ready

    for (int t = 0; t < NT_; ++t) {
        float* h2g = (t & 1) ? bufA : bufB;   // previous field; overwritten with hn

        // --- wait for this step's h1 tile in LDS ---
#ifdef HAVE_TDM
        if (tid < 32)
            __builtin_amdgcn_s_wait_tensorcnt(0);
#else
        {   // Fallback: cooperative device-scope load into LDS.
            const float* src = ((t & 1) ? bufB : bufA) + tile_off;
            for (int j = tid; j < TILE_ROWS * NX_; j += NTHREADS)
                lds_h1[j] = __hip_atomic_load(&src[j], __ATOMIC_RELAXED,
                                              __HIP_MEMORY_SCOPE_AGENT);
        }
#endif
        __syncthreads();

        const float xt = xsrc[b * NT_ + t];

        // --- 5-point stencil from LDS; hn written in place over h2 buffer ---
        int row = tid / NX_;              // 0 for tid<384... tid<256 so row=0
        int col = tid;                    // col within row
        #pragma unroll
        for (int i = 0; i < PTS_PER_THREAD; ++i) {
            int z = z0 + row;
            if (z >= 1 && z <= NZ_ - 2 && col >= 1 && col <= NX_ - 2) {
                float c   = lds_c2[row * NX_ + col];
                float cc  = lds_h1[(row + 1) * NX_ + col];
                float up  = lds_h1[(row    ) * NX_ + col];
                float dn  = lds_h1[(row + 2) * NX_ + col];
                float lf  = lds_h1[(row + 1) * NX_ + col - 1];
                float rt  = lds_h1[(row + 1) * NX_ + col + 1];
                size_t gp = batch_base + (size_t)(z + 1) * NX_ + col;
                float h2v = h2g[gp];                 // own point (same WG wrote it)
                float lap = (up + dn + lf + rt - 4.0f * cc) * inv_dh2;
                float hn  = 2.0f * cc - h2v + c * lap;
                if (z == sy && col == sx) hn += xt;  // source injection
                h2g[gp] = hn;
            }
            col += NTHREADS;                         // advance by 256 within 384-wide rows
            if (col >= NX_) { col -= NX_; ++row; }
        }

        // --- device-wide barrier (monotonic counter, zeroed by init kernel) ---
        __threadfence();
        if (tid == 0) {
            atomicAdd(barrier_cnt, 1u);
            const unsigned target = (unsigned)NWG * (unsigned)(t + 1);
            while (__hip_atomic_load(barrier_cnt, __ATOMIC_RELAXED,
                                     __HIP_MEMORY_SCOPE_AGENT) < target)
                __builtin_amdgcn_s_sleep(2);
        }
        __syncthreads();   // all LDS reads of step t complete beyond this point

#ifdef HAVE_TDM
        // --- kick off step t+1's tile DMA (reads the buffer just written) ---
        if (t + 1 < NT_ && tid < 32)
            tdm_issue_tile(lds_off, (uint64_t)(uintptr_t)(h2g + tile_off));
#endif

        // --- receiver gather from completed hn (device-scope, bypass stale WGP$) ---
        if (gtid < B_ * NREC_) {
            float v = __hip_atomic_load(&h2g[goff], __ATOMIC_RELAXED,
                                        __HIP_MEMORY_SCOPE_AGENT);
            out[(size_t)t * (B_ * NREC_) + gtid] = v;
        }
        // Step t+1 writes the OTHER buffer, so the gather and the in-flight TDM
        // (both readers of h2g) never conflict with the next step's writes.
    }
}

__global__ void wave_init_kernel(float* bufA, float* bufB, unsigned* cnt, int n) {
    int i = blockIdx.x * blockDim.x + threadIdx.x;
    if (i < n) { bufA[i] = 0.0f; bufB[i] = 0.0f; }
    if (i == 0) *cnt = 0u;
}

extern "C" void kernel_launch(void* const* d_in, const int* in_sizes, int n_in,
                              void* d_out, int out_size, void* d_ws, size_t ws_size,
                              hipStream_t stream) {
    const float* x     = (const float*)d_in[0];  // (B, NT)
    const float* vp    = (const float*)d_in[1];  // (NZ, NX)
    const int*   src_y = (const int*)d_in[2];
    const int*   src_x = (const int*)d_in[3];
    const int*   rec_y = (const int*)d_in[4];
    const int*   rec_x = (const int*)d_in[5];
    float*       out   = (float*)d_out;          // (NT, B, NREC)

    const int field_elems = B_ * NZP_ * NX_;     // 592896 floats per buffer
    float*    bufA = (float*)d_ws;
    float*    bufB = bufA + field_elems;
    unsigned* cnt  = (unsigned*)(bufB + field_elems);

    int init_blocks = (field_elems + NTHREADS - 1) / NTHREADS;
    wave_init_kernel<<<init_blocks, NTHREADS, 0, stream>>>(bufA, bufB, cnt, field_elems);

    wave_persist_kernel<<<NWG, NTHREADS, 0, stream>>>(
        x, vp, src_y, src_x, rec_y, rec_x, out, bufA, bufB, cnt);
}